// STEM_1614907703715
// MI455X (gfx1250) — compile-verified
//
#include <hip/hip_runtime.h>

// ---------------- problem constants ----------------
#define BB   64
#define CC   3
#define JJ   25
#define TT   256
#define CO   64
#define NPIX (BB*JJ*TT)        // 409600
#define SIGC 84
#define KTOT 168               // sp(84) + tp(84) fused K
#define EPSV 1e-5f

typedef float v2f __attribute__((ext_vector_type(2)));
typedef float v8f __attribute__((ext_vector_type(8)));

// ---------------- workspace layout (float offsets) ----------------
#define WS_ACONV ((size_t)0)                         // [NPIX][64] pre-BN relu(conv)
#define WS_APS   (WS_ACONV + (size_t)NPIX*CO)        // [NPIX][64] pre-BN relu(ps)
#define WS_WCAT  (WS_APS   + (size_t)NPIX*CO)        // [64][168] fused (sp|tp) folded weights, o-major
#define WS_BPS   (WS_WCAT + (size_t)CO*KTOT)         // [64] folded ps bias
#define WS_FUW   (WS_BPS + 64)                       // [64][128] fu_w with BN scales folded in
#define WS_FUB   (WS_FUW + (size_t)CO*128)           // [64] folded fu bias
#define WS_SCONV (WS_FUB + 64)                       // 6 x 64 stats (sum / sumsq) follow
#define WS_QCONV (WS_SCONV + 64)
#define WS_SPS   (WS_QCONV + 64)
#define WS_QPS   (WS_SPS + 64)
#define WS_SFU   (WS_QPS + 64)
#define WS_QFU   (WS_SFU + 64)

// ---------------- fp32 WMMA (16x16x4) ----------------
__device__ __forceinline__ v8f wmma4(v2f a, v2f b, v8f c) {
  // D = A(16x4,f32) * B(4x16,f32) + C(16x16,f32)
  return __builtin_amdgcn_wmma_f32_16x16x4_f32(false, a, false, b, (short)0, c, false, false);
}

// ---------------- level-3 signature of a 3-point path in R^4 ----------------
__device__ __forceinline__ void sig3_step(float S1[4], float S2[16], float S3[64],
                                          const float B1[4]) {
  float B2[16];
  #pragma unroll
  for (int a = 0; a < 4; ++a)
    #pragma unroll
    for (int b = 0; b < 4; ++b)
      B2[a*4+b] = 0.5f * B1[a] * B1[b];
  #pragma unroll
  for (int a = 0; a < 4; ++a)
    #pragma unroll
    for (int b = 0; b < 4; ++b) {
      float t2 = S2[a*4+b] + (1.0f/3.0f) * B2[a*4+b];
      #pragma unroll
      for (int c = 0; c < 4; ++c)
        S3[a*16+b*4+c] += t2 * B1[c] + S1[a] * B2[b*4+c];
    }
  #pragma unroll
  for (int a = 0; a < 4; ++a)
    #pragma unroll
    for (int b = 0; b < 4; ++b)
      S2[a*4+b] += B2[a*4+b] + S1[a] * B1[b];
  #pragma unroll
  for (int a = 0; a < 4; ++a) S1[a] += B1[a];
}

__device__ __forceinline__ void sig3_path(const float P0[4], const float P1[4],
                                          const float P2[4], float* __restrict__ outLds) {
  float I0[4], I1[4], I2[4];
  #pragma unroll
  for (int a = 0; a < 4; ++a) { I0[a] = P0[a]; I1[a] = P1[a]-P0[a]; I2[a] = P2[a]-P1[a]; }
  float S1[4], S2[16], S3[64];
  #pragma unroll
  for (int a = 0; a < 4; ++a) S1[a] = I0[a];
  #pragma unroll
  for (int a = 0; a < 4; ++a)
    #pragma unroll
    for (int b = 0; b < 4; ++b)
      S2[a*4+b] = 0.5f * I0[a] * I0[b];
  #pragma unroll
  for (int a = 0; a < 4; ++a)
    #pragma unroll
    for (int b = 0; b < 4; ++b)
      #pragma unroll
      for (int c = 0; c < 4; ++c)
        S3[a*16+b*4+c] = S2[a*4+b] * I0[c] * (1.0f/3.0f);
  sig3_step(S1, S2, S3, I1);
  sig3_step(S1, S2, S3, I2);
  #pragma unroll
  for (int a = 0; a < 4; ++a)  outLds[a]    = S1[a];
  #pragma unroll
  for (int i = 0; i < 16; ++i) outLds[4+i]  = S2[i];
  #pragma unroll
  for (int i = 0; i < 64; ++i) outLds[20+i] = S3[i];
}

// ---------------- k_pre: fuse ps matmul into the sig matmuls; zero stats ----------------
__global__ __launch_bounds__(256) void k_pre(const float* __restrict__ spW,
    const float* __restrict__ spb, const float* __restrict__ tpW,
    const float* __restrict__ tpb, const float* __restrict__ psw,
    const float* __restrict__ psb, float* __restrict__ ws) {
  int tid = threadIdx.x;
  for (int i = tid; i < 6*64; i += 256) ws[WS_SCONV + i] = 0.0f;
  for (int idx = tid; idx < CO*KTOT; idx += 256) {    // idx = o*168 + k
    int o = idx / KTOT;
    int k = idx - o*KTOT;
    float s = 0.0f;
    if (k < SIGC) {
      for (int c = 0; c < 64; ++c)
        s = fmaf(spW[k*64 + c], psw[o*128 + c], s);
    } else {
      int kk = k - SIGC;
      for (int c = 0; c < 64; ++c)
        s = fmaf(tpW[kk*64 + c], psw[o*128 + 64 + c], s);
    }
    ws[WS_WCAT + idx] = s;
  }
  if (tid < 64) {
    float s = psb[tid];
    for (int c = 0; c < 64; ++c) {
      s = fmaf(psw[tid*128 + c],      spb[c], s);
      s = fmaf(psw[tid*128 + 64 + c], tpb[c], s);
    }
    ws[WS_BPS + tid] = s;
  }
}

// ---------------- k_conv: relu(conv1x3 + b) -> ws, channel stats ----------------
__global__ __launch_bounds__(256) void k_conv(const float* __restrict__ x,
    const float* __restrict__ cw, const float* __restrict__ cb,
    float* __restrict__ ws) {
  __shared__ float ssum[CO], ssq[CO];
  int tid = threadIdx.x;
  int o  = tid & 63;
  int lp = tid >> 6;              // 0..3
  if (tid < CO) { ssum[tid] = 0.0f; ssq[tid] = 0.0f; }
  __syncthreads();
  float w[9];
  #pragma unroll
  for (int i = 0; i < 9; ++i) w[i] = cw[o*9 + i];   // (CO,C,1,3)
  float bias = cb[o];
  float* aconv = ws + WS_ACONV;
  float lsum = 0.0f, lsq = 0.0f;
  for (int pp = 0; pp < 4; ++pp) {
    int p = blockIdx.x*16 + pp*4 + lp;
    int b = p / (JJ*TT);
    int r = p - b*(JJ*TT);
    int j = r >> 8;
    int t = r & 255;
    float acc = bias;
    #pragma unroll
    for (int c = 0; c < 3; ++c) {
      const float* xr = x + (((size_t)b*3 + c)*JJ + j)*TT + t;
      float xm = (t > 0)    ? xr[-1] : 0.0f;   // zero padding
      float x0 = xr[0];
      float xp = (t < TT-1) ? xr[1]  : 0.0f;
      acc = fmaf(w[c*3+0], xm, acc);
      acc = fmaf(w[c*3+1], x0, acc);
      acc = fmaf(w[c*3+2], xp, acc);
    }
    float v = fmaxf(acc, 0.0f);
    aconv[(size_t)p*CO + o] = v;
    lsum += v; lsq += v*v;
  }
  atomicAdd(&ssum[o], lsum);
  atomicAdd(&ssq[o],  lsq);
  __syncthreads();
  if (tid < CO) {
    atomicAdd(&ws[WS_SCONV + tid], ssum[tid]);
    atomicAdd(&ws[WS_QCONV + tid], ssq[tid]);
  }
}

// ---------------- k_sig: signatures -> LDS -> fp32 WMMA GEMM (K=168 -> 64) ----------------
// block = 64 threads (2 waves) = 64 pixels; each wave owns 2 M-tiles sharing B operands.
__global__ __launch_bounds__(64) void k_sig(const float* __restrict__ x,
    const int* __restrict__ neighbors, float* __restrict__ ws) {
  __shared__ float feats[64*170];    // 64 pixels x 168 feats (sp|tp), stride 170
  __shared__ float wchunk[64*30];    // staged K-chunk of weights: [o][28], stride 30
  __shared__ float ssum[CO], ssq[CO];
  int tid = threadIdx.x;
  ssum[tid] = 0.0f; ssq[tid] = 0.0f;

  int p = blockIdx.x*64 + tid;
  int b = p / (JJ*TT);
  int r = p - b*(JJ*TT);
  int j = r >> 8;
  int t = r & 255;

  float P0[4], P1[4], P2[4];
  // spatial path: neighbor joints at time t, time-augmented with [0, .5, 1]
  int n0 = neighbors[j*3+0], n1 = neighbors[j*3+1], n2 = neighbors[j*3+2];
  P0[0] = 0.0f; P1[0] = 0.5f; P2[0] = 1.0f;
  #pragma unroll
  for (int c = 0; c < 3; ++c) {
    const float* xb = x + (((size_t)b*3 + c)*JJ)*TT + t;
    P0[c+1] = xb[n0*TT];
    P1[c+1] = xb[n1*TT];
    P2[c+1] = xb[n2*TT];
  }
  sig3_path(P0, P1, P2, &feats[tid*170]);

  // temporal path: (t-1, t, t+1) edge-clamped
  int tm = (t > 0)    ? t-1 : 0;
  int t2 = (t < TT-1) ? t+1 : TT-1;
  P0[0] = 0.0f; P1[0] = 0.5f; P2[0] = 1.0f;
  #pragma unroll
  for (int c = 0; c < 3; ++c) {
    const float* xb = x + (((size_t)b*3 + c)*JJ + j)*TT;
    P0[c+1] = xb[tm]; P1[c+1] = xb[t]; P2[c+1] = xb[t2];
  }
  sig3_path(P0, P1, P2, &feats[tid*170 + 84]);

  // ---- WMMA phase: D[2 x 16px x 64ch] accumulated over K=168 in 6 chunks of 28 ----
  const float* Wcat = ws + WS_WCAT;  // [o][168]
  const float* bps  = ws + WS_BPS;
  float* aps = ws + WS_APS;
  int lane = tid & 31;
  int wave = tid >> 5;
  int mcol = lane & 15;              // A row / B,C,D column within tile
  int kh   = (lane >> 4) << 1;       // 0 for lanes 0-15, 2 for lanes 16-31
  const float* fA0 = &feats[(wave*32 +      mcol)*170];
  const float* fA1 = &feats[(wave*32 + 16 + mcol)*170];

  v8f acc0[4] = {};
  v8f acc1[4] = {};
  for (int ch = 0; ch < 6; ++ch) {
    __syncthreads();                 // feats ready (ch==0) / previous chunk consumed
    {
      const float* src = &Wcat[tid*KTOT + ch*28];
      float* dst = &wchunk[tid*30];
      #pragma unroll
      for (int i = 0; i < 28; i += 2)
        *(v2f*)&dst[i] = *(const v2f*)&src[i];
    }
    __syncthreads();
    #pragma unroll
    for (int ks = 0; ks < 7; ++ks) {
      int kl = ks*4 + kh;            // k within chunk
      int kg = ch*28 + kl;           // k within feats
      v2f a0 = *(const v2f*)&fA0[kg];
      v2f a1 = *(const v2f*)&fA1[kg];
      v2f b0 = *(const v2f*)&wchunk[( 0 + mcol)*30 + kl];
      v2f b1 = *(const v2f*)&wchunk[(16 + mcol)*30 + kl];
      v2f b2 = *(const v2f*)&wchunk[(32 + mcol)*30 + kl];
      v2f b3 = *(const v2f*)&wchunk[(48 + mcol)*30 + kl];
      acc0[0] = wmma4(a0, b0, acc0[0]);
      acc0[1] = wmma4(a0, b1, acc0[1]);
      acc0[2] = wmma4(a0, b2, acc0[2]);
      acc0[3] = wmma4(a0, b3, acc0[3]);
      acc1[0] = wmma4(a1, b0, acc1[0]);
      acc1[1] = wmma4(a1, b1, acc1[1]);
      acc1[2] = wmma4(a1, b2, acc1[2]);
      acc1[3] = wmma4(a1, b3, acc1[3]);
    }
  }

  // ---- epilogue: bias, relu, store pre-BN activations + channel stats ----
  #pragma unroll
  for (int mt = 0; mt < 2; ++mt) {
    int p0 = blockIdx.x*64 + wave*32 + mt*16;
    #pragma unroll
    for (int n = 0; n < 4; ++n) {
      int o = n*16 + mcol;
      float bias = bps[o];
      float s = 0.0f, q = 0.0f;
      #pragma unroll
      for (int v = 0; v < 8; ++v) {
        int m = (lane < 16) ? v : (v + 8);
        float val = (mt == 0) ? acc0[n][v] : acc1[n][v];
        val = fmaxf(val + bias, 0.0f);
        aps[(size_t)(p0 + m)*CO + o] = val;
        s += val; q += val*val;
      }
      atomicAdd(&ssum[o], s);
      atomicAdd(&ssq[o],  q);
    }
  }
  __syncthreads();
  atomicAdd(&ws[WS_SPS + tid], ssum[tid]);
  atomicAdd(&ws[WS_QPS + tid], ssq[tid]);
}

// ---------------- k_bnprep: fold conv-BN & ps-BN into fu_w / fu_b ----------------
__global__ __launch_bounds__(256) void k_bnprep(const float* __restrict__ rg,
    const float* __restrict__ rb, const float* __restrict__ pg,
    const float* __restrict__ pb, const float* __restrict__ fw,
    const float* __restrict__ fbv, float* __restrict__ ws) {
  __shared__ float scale[128], off[128];
  int tid = threadIdx.x;
  const float invN = 1.0f / (float)NPIX;
  if (tid < 128) {
    float s, q, g, be;
    if (tid < 64) { s = ws[WS_SCONV+tid]; q = ws[WS_QCONV+tid]; g = rg[tid]; be = rb[tid]; }
    else { int c = tid-64; s = ws[WS_SPS+c]; q = ws[WS_QPS+c]; g = pg[c]; be = pb[c]; }
    float m   = s*invN;
    float var = q*invN - m*m;
    float rs  = rsqrtf(var + EPSV);
    scale[tid] = g*rs;
    off[tid]   = be - m*g*rs;
  }
  __syncthreads();
  for (int idx = tid; idx < CO*128; idx += 256) {   // idx = o*128 + k
    int k = idx & 127;
    ws[WS_FUW + idx] = fw[idx] * scale[k];
  }
  if (tid < 64) {
    float s = fbv[tid];
    for (int k = 0; k < 128; ++k) s = fmaf(fw[tid*128 + k], off[k], s);
    ws[WS_FUB + tid] = s;
  }
}

// ---------------- k_fu: WMMA fusion GEMM (K=128 -> 64), relu, pre-BN out + stats ----------------
// block = 64 threads = 64 pixels; fu_w staged fully in LDS, activations staged per-half.
__global__ __launch_bounds__(64) void k_fu(float* __restrict__ ws, float* __restrict__ out) {
  __shared__ float wlds[64*130];     // [o][128], stride 130 (conflict-free)
  __shared__ float xbuf[64*66];      // [pixel][64] activation half, stride 66
  __shared__ float ssum[CO], ssq[CO];
  int tid = threadIdx.x;
  ssum[tid] = 0.0f; ssq[tid] = 0.0f;

  const float* ac = ws + WS_ACONV;
  const float* ap = ws + WS_APS;
  const float* Wf = ws + WS_FUW;
  const float* fb = ws + WS_FUB;

  // stage folded weights (row o = tid)
  {
    const float* src = &Wf[tid*128];
    float* dst = &wlds[tid*130];
    #pragma unroll
    for (int i = 0; i < 128; i += 2)
      *(v2f*)&dst[i] = *(const v2f*)&src[i];
  }
  // stage conv-half activations (pixel = tid)
  {
    const float* src = ac + (size_t)(blockIdx.x*64 + tid)*CO;
    float* dst = &xbuf[tid*66];
    #pragma unroll
    for (int i = 0; i < 64; i += 2)
      *(v2f*)&dst[i] = *(const v2f*)&src[i];
  }
  __syncthreads();

  int lane = tid & 31;
  int wave = tid >> 5;
  int mcol = lane & 15;
  int kh   = (lane >> 4) << 1;
  int rowA0 = (wave*32 +      mcol)*66;
  int rowA1 = (wave*32 + 16 + mcol)*66;

  v8f acc0[4] = {};
  v8f acc1[4] = {};
  // conv half (weights k = 0..63)
  #pragma unroll
  for (int ks = 0; ks < 16; ++ks) {
    int kl = ks*4 + kh;
    v2f a0 = *(const v2f*)&xbuf[rowA0 + kl];
    v2f a1 = *(const v2f*)&xbuf[rowA1 + kl];
    v2f b0 = *(const v2f*)&wlds[( 0 + mcol)*130 + kl];
    v2f b1 = *(const v2f*)&wlds[(16 + mcol)*130 + kl];
    v2f b2 = *(const v2f*)&wlds[(32 + mcol)*130 + kl];
    v2f b3 = *(const v2f*)&wlds[(48 + mcol)*130 + kl];
    acc0[0] = wmma4(a0, b0, acc0[0]);
    acc0[1] = wmma4(a0, b1, acc0[1]);
    acc0[2] = wmma4(a0, b2, acc0[2]);
    acc0[3] = wmma4(a0, b3, acc0[3]);
    acc1[0] = wmma4(a1, b0, acc1[0]);
    acc1[1] = wmma4(a1, b1, acc1[1]);
    acc1[2] = wmma4(a1, b2, acc1[2]);
    acc1[3] = wmma4(a1, b3, acc1[3]);
  }
  // swap in ps-half activations
  __syncthreads();
  {
    const float* src = ap + (size_t)(blockIdx.x*64 + tid)*CO;
    float* dst = &xbuf[tid*66];
    #pragma unroll
    for (int i = 0; i < 64; i += 2)
      *(v2f*)&dst[i] = *(const v2f*)&src[i];
  }
  __syncthreads();
  // ps half (weights k = 64..127)
  #pragma unroll
  for (int ks = 0; ks < 16; ++ks) {
    int kl = ks*4 + kh;
    v2f a0 = *(const v2f*)&xbuf[rowA0 + kl];
    v2f a1 = *(const v2f*)&xbuf[rowA1 + kl];
    v2f b0 = *(const v2f*)&wlds[( 0 + mcol)*130 + 64 + kl];
    v2f b1 = *(const v2f*)&wlds[(16 + mcol)*130 + 64 + kl];
    v2f b2 = *(const v2f*)&wlds[(32 + mcol)*130 + 64 + kl];
    v2f b3 = *(const v2f*)&wlds[(48 + mcol)*130 + 64 + kl];
    acc0[0] = wmma4(a0, b0, acc0[0]);
    acc0[1] = wmma4(a0, b1, acc0[1]);
    acc0[2] = wmma4(a0, b2, acc0[2]);
    acc0[3] = wmma4(a0, b3, acc0[3]);
    acc1[0] = wmma4(a1, b0, acc1[0]);
    acc1[1] = wmma4(a1, b1, acc1[1]);
    acc1[2] = wmma4(a1, b2, acc1[2]);
    acc1[3] = wmma4(a1, b3, acc1[3]);
  }

  // epilogue: bias, relu, scatter to (B,CO,J,T) output + stats
  #pragma unroll
  for (int mt = 0; mt < 2; ++mt) {
    int p0 = blockIdx.x*64 + wave*32 + mt*16;
    #pragma unroll
    for (int n = 0; n < 4; ++n) {
      int o = n*16 + mcol;
      float bias = fb[o];
      float s = 0.0f, q = 0.0f;
      #pragma unroll
      for (int v = 0; v < 8; ++v) {
        int m = (lane < 16) ? v : (v + 8);
        int p = p0 + m;
        float val = (mt == 0) ? acc0[n][v] : acc1[n][v];
        val = fmaxf(val + bias, 0.0f);
        int b = p / (JJ*TT);
        int r = p - b*(JJ*TT);
        int j = r >> 8;
        int t = r & 255;
        out[(((size_t)b*CO + o)*JJ + j)*TT + t] = val;
        s += val; q += val*val;
      }
      atomicAdd(&ssum[o], s);
      atomicAdd(&ssq[o],  q);
    }
  }
  __syncthreads();
  atomicAdd(&ws[WS_SFU + tid], ssum[tid]);
  atomicAdd(&ws[WS_QFU + tid], ssq[tid]);
}

// ---------------- k_final: in-place final BN of d_out ----------------
__global__ __launch_bounds__(256) void k_final(const float* __restrict__ fg,
    const float* __restrict__ fbeta, const float* __restrict__ ws,
    float* __restrict__ out) {
  int idx = blockIdx.x*256 + threadIdx.x;           // grid covers exactly NPIX*CO
  int o = (idx / (JJ*TT)) & 63;
  const float invN = 1.0f / (float)NPIX;
  float s = ws[WS_SFU + o], q = ws[WS_QFU + o];
  float m   = s*invN;
  float var = q*invN - m*m;
  float rs  = rsqrtf(var + EPSV);
  float v = out[idx];
  out[idx] = (v - m)*rs*fg[o] + fbeta[o];
}

// ---------------- host launcher ----------------
extern "C" void kernel_launch(void* const* d_in, const int* in_sizes, int n_in,
                              void* d_out, int out_size, void* d_ws, size_t ws_size,
                              hipStream_t stream) {
  const float* x      = (const float*)d_in[0];
  const float* conv_w = (const float*)d_in[1];
  const float* conv_b = (const float*)d_in[2];
  const float* raw_g  = (const float*)d_in[3];
  const float* raw_b  = (const float*)d_in[4];
  const float* sp_W   = (const float*)d_in[5];
  const float* sp_b   = (const float*)d_in[6];
  const float* tp_W   = (const float*)d_in[7];
  const float* tp_b   = (const float*)d_in[8];
  const float* ps_w   = (const float*)d_in[9];
  const float* ps_b   = (const float*)d_in[10];
  const float* ps_g   = (const float*)d_in[11];
  const float* ps_be  = (const float*)d_in[12];
  const float* fu_w   = (const float*)d_in[13];
  const float* fu_b   = (const float*)d_in[14];
  const float* fu_g   = (const float*)d_in[15];
  const float* fu_be  = (const float*)d_in[16];
  const int*   neigh  = (const int*)d_in[17];
  float* ws  = (float*)d_ws;
  float* out = (float*)d_out;
  (void)in_sizes; (void)n_in; (void)out_size; (void)ws_size;

  k_pre   <<<1,        256, 0, stream>>>(sp_W, sp_b, tp_W, tp_b, ps_w, ps_b, ws);
  k_conv  <<<NPIX/16,  256, 0, stream>>>(x, conv_w, conv_b, ws);
  k_sig   <<<NPIX/64,   64, 0, stream>>>(x, neigh, ws);
  k_bnprep<<<1,        256, 0, stream>>>(raw_g, raw_b, ps_g, ps_be, fu_w, fu_b, ws);
  k_fu    <<<NPIX/64,   64, 0, stream>>>(ws, out);
  k_final <<<(NPIX*CO)/256, 256, 0, stream>>>(fu_g, fu_be, ws, out);
}